// GCN_N_76630806495980
// MI455X (gfx1250) — compile-verified
//
#include <hip/hip_runtime.h>
#include <hip/hip_bf16.h>

typedef __attribute__((ext_vector_type(2))) float v2f;
typedef __attribute__((ext_vector_type(8))) float v8f;

#define N_NODES  100000
#define N_EDGES  1600000
#define N_GRAPHS 2048
#define F_IN     9
#define DIM      32

static __device__ __forceinline__ void atomAddF(float* p, float v) {
    // native global_atomic_add_f32 path
    unsafeAtomicAdd(p, v);
}

__global__ void k_zero(float* __restrict__ p, long n) {
    long i = (long)blockIdx.x * blockDim.x + threadIdx.x;
    if (i < n) p[i] = 0.0f;
}

__global__ void k_deg(const int* __restrict__ dst, float* __restrict__ deg) {
    int e = blockIdx.x * blockDim.x + threadIdx.x;
    if (e < N_EDGES) atomAddF(&deg[dst[e]], 1.0f);
}

__global__ void k_dis(float* __restrict__ d) {
    int i = blockIdx.x * blockDim.x + threadIdx.x;
    if (i < N_NODES) d[i] = rsqrtf(d[i] + 2.0f);
}

// Layer-1 GEMM: [N,9] @ [9,32] -> [N,32]  (K=9, VALU path)
__global__ void k_gemm1(const float* __restrict__ x, const float* __restrict__ W,
                        float* __restrict__ xw) {
    int t = blockIdx.x * blockDim.x + threadIdx.x;
    if (t >= N_NODES * DIM) return;
    int node = t >> 5, d = t & 31;
    float acc = 0.0f;
    #pragma unroll
    for (int k = 0; k < F_IN; ++k)
        acc += x[node * F_IN + k] * W[k * DIM + d];
    xw[t] = acc;
}

// Shared-weight GEMM: [N,32] @ [32,32] -> [N,32] via V_WMMA_F32_16X16X4_F32.
// One wave32 handles a 16-row tile. K=32 -> 8 chunks of 4; 2 column halves.
// A 16x4 f32 layout (ISA 7.12.2): lane m (0..15) holds K=+0,+1; lanes 16..31 K=+2,+3.
// B 4x16 f32: VGPR v, lanes 0..15 hold row K=v, lanes 16..31 hold row K=v+2.
// C/D 16x16 f32: VGPR r, lanes 0..15 -> row r, lanes 16..31 -> row r+8.
__global__ void k_gemm32(const float* __restrict__ h, const float* __restrict__ W,
                         float* __restrict__ xw) {
    int wave = (blockIdx.x * blockDim.x + threadIdx.x) >> 5;
    int lane = threadIdx.x & 31;
    const int ntiles = N_NODES / 16;     // 6250, exact
    if (wave >= ntiles) return;

    int m    = lane & 15;                // row-in-tile (A) / column (B, C/D)
    int khi  = (lane >> 4) << 1;         // 0 for lanes 0..15, 2 for lanes 16..31
    int row  = wave * 16 + m;
    const float* hrow = h + (size_t)row * DIM;

    v8f c0 = {}; // columns 0..15
    v8f c1 = {}; // columns 16..31
    #pragma unroll
    for (int kb = 0; kb < DIM; kb += 4) {
        v2f a;
        a.x = hrow[kb + khi];
        a.y = hrow[kb + khi + 1];
        v2f b0, b1;
        b0.x = W[(kb + khi)     * DIM + m];
        b0.y = W[(kb + khi + 1) * DIM + m];
        b1.x = W[(kb + khi)     * DIM + m + 16];
        b1.y = W[(kb + khi + 1) * DIM + m + 16];
        c0 = __builtin_amdgcn_wmma_f32_16x16x4_f32(false, a, false, b0,
                                                   (short)0, c0, false, false);
        c1 = __builtin_amdgcn_wmma_f32_16x16x4_f32(false, a, false, b1,
                                                   (short)0, c1, false, false);
    }

    int rbase = wave * 16 + ((lane >> 4) << 3); // upper half-lanes own rows +8
    #pragma unroll
    for (int r = 0; r < 8; ++r) {
        xw[(size_t)(rbase + r) * DIM + m]      = c0[r];
        xw[(size_t)(rbase + r) * DIM + m + 16] = c1[r];
    }
}

// Edge scatter: agg[dst] += dis[dst]*dis[src] * xw[src]  (32 lanes = one edge row)
__global__ void k_edge(const int* __restrict__ src, const int* __restrict__ dst,
                       const float* __restrict__ dis, const float* __restrict__ xw,
                       float* __restrict__ agg) {
    long t = (long)blockIdx.x * blockDim.x + threadIdx.x;
    if (t >= (long)N_EDGES * DIM) return;
    int e = (int)(t >> 5);
    int d = (int)(t & 31);
    int s = src[e], q = dst[e];
    float coeff = dis[s] * dis[q];
    atomAddF(&agg[(size_t)q * DIM + d], coeff * xw[(size_t)s * DIM + d]);
    // warm L2 with an upcoming gather row (global_prefetch_b8; cacheline covers full row)
    if (d == 0 && e + 256 < N_EDGES)
        __builtin_prefetch(&xw[(size_t)src[e + 256] * DIM], 0, 1);
}

// h = relu(agg + 2*dis^2*xw + b)
__global__ void k_self_relu(const float* __restrict__ agg, const float* __restrict__ xw,
                            const float* __restrict__ dis, const float* __restrict__ bias,
                            float* __restrict__ hout) {
    int t = blockIdx.x * blockDim.x + threadIdx.x;
    if (t >= N_NODES * DIM) return;
    int i = t >> 5, d = t & 31;
    float ds = dis[i];
    float v = agg[t] + 2.0f * ds * ds * xw[t] + bias[d];
    hout[t] = v > 0.0f ? v : 0.0f;
}

__global__ void k_pool(const float* __restrict__ h, const int* __restrict__ batch,
                       float* __restrict__ g) {
    int t = blockIdx.x * blockDim.x + threadIdx.x;
    if (t >= N_NODES * DIM) return;
    int i = t >> 5, d = t & 31;
    atomAddF(&g[(size_t)batch[i] * DIM + d], h[t]);
}

__global__ void k_out(const float* __restrict__ g, const float* __restrict__ Wout,
                      const float* __restrict__ bout, float* __restrict__ out) {
    int i = blockIdx.x * blockDim.x + threadIdx.x;
    if (i >= N_GRAPHS) return;
    float acc = bout[0];
    #pragma unroll
    for (int k = 0; k < DIM; ++k)
        acc += g[i * DIM + k] * Wout[k];
    out[i] = acc;
}

extern "C" void kernel_launch(void* const* d_in, const int* in_sizes, int n_in,
                              void* d_out, int out_size, void* d_ws, size_t ws_size,
                              hipStream_t stream) {
    const float* x    = (const float*)d_in[0];
    const float* W1   = (const float*)d_in[1];
    const float* b1   = (const float*)d_in[2];
    const float* Wn   = (const float*)d_in[3];
    const float* bn   = (const float*)d_in[4];
    const float* Wout = (const float*)d_in[5];
    const float* bout = (const float*)d_in[6];
    const int*   src  = (const int*)d_in[7];
    const int*   dst  = (const int*)d_in[8];
    const int*   batch= (const int*)d_in[9];
    float* out = (float*)d_out;

    float* ws  = (float*)d_ws;
    float* dis = ws;                                     // N_NODES
    float* xw  = dis + N_NODES;                          // N_NODES*DIM
    float* agg = xw  + (size_t)N_NODES * DIM;            // N_NODES*DIM
    float* h   = agg + (size_t)N_NODES * DIM;            // N_NODES*DIM
    float* g   = h   + (size_t)N_NODES * DIM;            // N_GRAPHS*DIM

    const int B = 256;
    auto nb = [](long n) { return (int)((n + 255) / 256); };
    const long ND = (long)N_NODES * DIM;
    const long ED = (long)N_EDGES * DIM;

    // degree -> dis = rsqrt(deg + 2)
    k_zero<<<nb(N_NODES), B, 0, stream>>>(dis, N_NODES);
    k_deg <<<nb(N_EDGES), B, 0, stream>>>(dst, dis);
    k_dis <<<nb(N_NODES), B, 0, stream>>>(dis);

    // layer 1 (W1, b1)
    k_gemm1<<<nb(ND), B, 0, stream>>>(x, W1, xw);
    k_zero <<<nb(ND), B, 0, stream>>>(agg, ND);
    k_edge <<<nb(ED), B, 0, stream>>>(src, dst, dis, xw, agg);
    k_self_relu<<<nb(ND), B, 0, stream>>>(agg, xw, dis, b1, h);

    // layers 2..3 (shared Wn, bn) — WMMA f32 GEMM
    const int gemm_threads = (N_NODES / 16) * 32; // one wave per 16-row tile
    for (int l = 0; l < 2; ++l) {
        k_gemm32<<<nb(gemm_threads), B, 0, stream>>>(h, Wn, xw);
        k_zero  <<<nb(ND), B, 0, stream>>>(agg, ND);
        k_edge  <<<nb(ED), B, 0, stream>>>(src, dst, dis, xw, agg);
        k_self_relu<<<nb(ND), B, 0, stream>>>(agg, xw, dis, bn, h);
    }

    // global_add_pool + output head
    k_zero<<<nb((long)N_GRAPHS * DIM), B, 0, stream>>>(g, (long)N_GRAPHS * DIM);
    k_pool<<<nb(ND), B, 0, stream>>>(h, batch, g);
    k_out <<<nb(N_GRAPHS), B, 0, stream>>>(g, Wout, bout, out);
}